// Seq2Seq_72138270704176
// MI455X (gfx1250) — compile-verified
//
#include <hip/hip_runtime.h>
#include <hip/hip_bf16.h>
#include <math.h>

// ---------------- problem constants ----------------
#define BB      4        // batch
#define BP      16       // batch padded to one WMMA tile row-count
#define TT      128      // encoder time steps
#define HH      1024     // hidden
#define EE      512      // embedding
#define AA      512      // attention size
#define VV      16000    // vocab
#define SS      512      // decoder steps
#define H3      (3*HH)   // 3072

typedef __attribute__((ext_vector_type(2))) float v2f;
typedef __attribute__((ext_vector_type(8))) float v8f;

// =====================================================================
// Core f32 WMMA tile routine: one wave computes a 16x16 tile of
// C[M,N] = act( A[M,K] @ W[N,K]^T + bias[N] ) using V_WMMA_F32_16X16X4_F32.
//   A frag (16x4, MxK): lane l -> m=l&15, k = 2*(l>>4) + {0,1}
//   B frag (4x16, KxN): lane l -> n=l&15, k = 2*(l>>4) + {0,1}, B[k][n]=W[n*K+k]
//   C/D (16x16): vgpr i, lanes0-15 -> m=i, lanes16-31 -> m=8+i, n=lane&15
// A must be padded to 16*tilesM rows (pad rows zero) so loads are unguarded.
// =====================================================================
__device__ __forceinline__ void wmma_tile(const float* __restrict__ A,
                                          const float* __restrict__ W,
                                          const float* __restrict__ bias,
                                          float* __restrict__ C,
                                          int M, int N, int K, int act,
                                          int tm, int tn, int lane)
{
    const int half = lane >> 4;     // 0: k=0,1   1: k=2,3
    const int mr   = lane & 15;

    const float* Arow = A + (size_t)(tm * 16 + mr) * K + half * 2;
    const float* Wrow = W + (size_t)(tn * 16 + mr) * K + half * 2;

    v8f acc = {};
#pragma unroll 8
    for (int k0 = 0; k0 < K; k0 += 4) {
        v2f a = *(const v2f*)(Arow + k0);
        v2f b = *(const v2f*)(Wrow + k0);
        acc = __builtin_amdgcn_wmma_f32_16x16x4_f32(
            false, a, false, b, (short)0, acc, false, false);
    }

    const int col = tn * 16 + mr;
    const float bv = bias[col];
#pragma unroll
    for (int i = 0; i < 8; ++i) {
        int row = tm * 16 + i + half * 8;
        if (row < M) {
            float v = acc[i] + bv;
            if (act) v = tanhf(v);
            C[(size_t)row * N + col] = v;
        }
    }
}

// Generic single GEMM.
__global__ void wmma_gemm_xwt(const float* __restrict__ A,
                              const float* __restrict__ W,
                              const float* __restrict__ bias,
                              float* __restrict__ C,
                              int M, int N, int K, int act)
{
    const int wavesPerBlock = blockDim.x >> 5;
    const int gw     = blockIdx.x * wavesPerBlock + (threadIdx.x >> 5);
    const int tilesN = N >> 4;
    const int tilesM = (M + 15) >> 4;
    if (gw >= tilesM * tilesN) return;
    wmma_tile(A, W, bias, C, M, N, K, act,
              gw / tilesN, gw % tilesN, threadIdx.x & 31);
}

// Two independent same-shape GEMMs in a single launch (decoder gi & gh).
__global__ void wmma_gemm_dual(const float* __restrict__ A0,
                               const float* __restrict__ W0,
                               const float* __restrict__ bias0,
                               float* __restrict__ C0,
                               const float* __restrict__ A1,
                               const float* __restrict__ W1,
                               const float* __restrict__ bias1,
                               float* __restrict__ C1,
                               int M, int N, int K)
{
    const int wavesPerBlock = blockDim.x >> 5;
    const int gw     = blockIdx.x * wavesPerBlock + (threadIdx.x >> 5);
    const int tilesN = N >> 4;
    const int tilesM = (M + 15) >> 4;
    const int tiles  = tilesM * tilesN;
    if (gw >= 2 * tiles) return;
    const int lane = threadIdx.x & 31;
    if (gw < tiles)
        wmma_tile(A0, W0, bias0, C0, M, N, K, 0,
                  gw / tilesN, gw % tilesN, lane);
    else {
        int g = gw - tiles;
        wmma_tile(A1, W1, bias1, C1, M, N, K, 0,
                  g / tilesN, g % tilesN, lane);
    }
}

// =====================================================================
// Zero the padded state buffers once (pad rows must stay zero so the
// WMMA A-fragments can load unguarded).
// =====================================================================
__global__ void init_state_kernel(float* h, float* x_cat, float* merged,
                                  float* attn_vec)
{
    int i = blockIdx.x * blockDim.x + threadIdx.x;   // covers 16*2048
    if (i < BP * HH)     h[i]        = 0.f;
    if (i < BP * HH)     x_cat[i]    = 0.f;
    if (i < BP * 2 * HH) merged[i]   = 0.f;
    if (i < BP * AA)     attn_vec[i] = 0.f;
}

// =====================================================================
// Encoder embedding gather: emb_x[(b*T+t)*E + e] = enc_emb[inputs[b,t]*E + e]
// =====================================================================
__global__ void enc_embed_kernel(const int* __restrict__ inputs,
                                 const float* __restrict__ enc_emb,
                                 float* __restrict__ emb_x)
{
    int idx = blockIdx.x * blockDim.x + threadIdx.x;   // B*T*E
    if (idx >= BB * TT * EE) return;
    int bt = idx >> 9;          // /512
    int e  = idx & (EE - 1);
    emb_x[idx] = enc_emb[(size_t)inputs[bt] * EE + e];
}

// =====================================================================
// GRU gate math (standalone version for the encoder scan).
// =====================================================================
__device__ __forceinline__ float gru_gate_elem(float ir, float iz, float in,
                                               float hr, float hz, float hn,
                                               float hprev)
{
    float r = 1.f / (1.f + __expf(-(ir + hr)));
    float z = 1.f / (1.f + __expf(-(iz + hz)));
    float n = tanhf(in + r * hn);
    return (1.f - z) * n + z * hprev;
}

__global__ void gru_gate_kernel(const float* __restrict__ gi, int giStride,
                                const float* __restrict__ gh, int ghStride,
                                float* __restrict__ h,
                                float* __restrict__ aux, int auxStride)
{
    int idx = blockIdx.x * blockDim.x + threadIdx.x;  // B*H
    if (idx >= BB * HH) return;
    int b = idx >> 10;
    int j = idx & (HH - 1);

    float hnew = gru_gate_elem(gi[(size_t)b * giStride + j],
                               gi[(size_t)b * giStride + HH + j],
                               gi[(size_t)b * giStride + 2 * HH + j],
                               gh[(size_t)b * ghStride + j],
                               gh[(size_t)b * ghStride + HH + j],
                               gh[(size_t)b * ghStride + 2 * HH + j],
                               h[(size_t)b * HH + j]);
    h[(size_t)b * HH + j]          = hnew;
    aux[(size_t)b * auxStride + j] = hnew;
}

// =====================================================================
// Fused decoder gate + attention. One block (1024 threads = H) per batch:
//  - h_new = GRU gate(gi, gh, h): written to h (global), merged[:,H:], LDS
//  - score[t] = dot(enc_outputs[b,t,:], h_new) masked by inputs!=0
//  - softmax over t, weights -> d_out attention section
//  - context -> merged[:, 0:H]
// =====================================================================
__global__ void gate_attention_kernel(const float* __restrict__ gi,
                                      const float* __restrict__ gh,
                                      float* __restrict__ h,
                                      const int* __restrict__ inputs,
                                      const float* __restrict__ enc_outputs,
                                      float* __restrict__ merged,
                                      float* __restrict__ w_out) // + step offset
{
    __shared__ float s_h[HH];
    __shared__ float s_score[TT];
    __shared__ float s_inv;

    const int b    = blockIdx.x;
    const int tid  = threadIdx.x;       // 0..1023 == hidden index
    const int wave = tid >> 5;
    const int lane = tid & 31;

    // ---- GRU gate for element (b, tid) ----
    {
        float hnew = gru_gate_elem(gi[(size_t)b * H3 + tid],
                                   gi[(size_t)b * H3 + HH + tid],
                                   gi[(size_t)b * H3 + 2 * HH + tid],
                                   gh[(size_t)b * H3 + tid],
                                   gh[(size_t)b * H3 + HH + tid],
                                   gh[(size_t)b * H3 + 2 * HH + tid],
                                   h[(size_t)b * HH + tid]);
        h[(size_t)b * HH + tid]                = hnew;  // for next step's GEMM
        merged[(size_t)b * (2 * HH) + HH + tid] = hnew; // [context | h]
        s_h[tid] = hnew;
    }
    __syncthreads();

    const float* eo_b = enc_outputs + (size_t)b * TT * HH;

    // ---- scores: wave w handles t = w, w+32, w+64, w+96 ----
    for (int t = wave; t < TT; t += 32) {
        const float* row = eo_b + (size_t)t * HH;
        float p = 0.f;
        for (int i = lane; i < HH; i += 32) p += row[i] * s_h[i];
        for (int off = 16; off; off >>= 1) p += __shfl_xor(p, off);
        if (lane == 0)
            s_score[t] = (inputs[b * TT + t] != 0) ? p : -1e30f;
    }
    __syncthreads();

    // ---- softmax (cheap serial pass over 128) ----
    if (tid == 0) {
        float m = -INFINITY;
        for (int t = 0; t < TT; ++t) m = fmaxf(m, s_score[t]);
        float s = 0.f;
        for (int t = 0; t < TT; ++t) {
            float e = __expf(s_score[t] - m);
            s_score[t] = e;
            s += e;
        }
        s_inv = 1.f / s;
    }
    __syncthreads();
    if (tid < TT) {
        float wv = s_score[tid] * s_inv;
        s_score[tid] = wv;
        w_out[b * TT + tid] = wv;
    }
    __syncthreads();

    // ---- context: each thread owns one hidden element ----
    float c = 0.f;
    for (int t = 0; t < TT; ++t) c += s_score[t] * eo_b[(size_t)t * HH + tid];
    merged[(size_t)b * (2 * HH) + tid] = c;
}

// =====================================================================
// Argmax over this step's logits (first-index tie-break like jnp.argmax),
// then build next step's x_cat = [dec_emb[id] | attn_vec].
// =====================================================================
__global__ void argmax_xcat_kernel(const float* __restrict__ logits, // + s*B*V
                                   const float* __restrict__ dec_emb,
                                   const float* __restrict__ attn_vec,
                                   float* __restrict__ x_cat,
                                   int* __restrict__ ids)
{
    __shared__ float sv[256];
    __shared__ int   si[256];
    const int b   = blockIdx.x;
    const int tid = threadIdx.x;  // 256
    const float* lb = logits + (size_t)b * VV;

    float best = -INFINITY; int bi = 0;
    for (int i = tid; i < VV; i += 256) {
        float v = lb[i];
        if (v > best) { best = v; bi = i; }
    }
    sv[tid] = best; si[tid] = bi;
    __syncthreads();
    for (int s = 128; s; s >>= 1) {
        if (tid < s) {
            if (sv[tid + s] > sv[tid] ||
                (sv[tid + s] == sv[tid] && si[tid + s] < si[tid])) {
                sv[tid] = sv[tid + s]; si[tid] = si[tid + s];
            }
        }
        __syncthreads();
    }
    const int id = si[0];
    if (tid == 0) ids[b] = id;

    for (int j = tid; j < EE; j += 256) {
        x_cat[(size_t)b * (EE + AA) + j]      = dec_emb[(size_t)id * EE + j];
        x_cat[(size_t)b * (EE + AA) + EE + j] = attn_vec[(size_t)b * AA + j];
    }
}

// =====================================================================
// host orchestration
// =====================================================================
extern "C" void kernel_launch(void* const* d_in, const int* in_sizes, int n_in,
                              void* d_out, int out_size, void* d_ws, size_t ws_size,
                              hipStream_t stream)
{
    const int*   inputs  = (const int*)  d_in[0];
    const float* enc_emb = (const float*)d_in[1];
    const float* enc_Wih = (const float*)d_in[2];
    const float* enc_Whh = (const float*)d_in[3];
    const float* enc_bih = (const float*)d_in[4];
    const float* enc_bhh = (const float*)d_in[5];
    const float* dec_emb = (const float*)d_in[6];
    const float* dec_Wih = (const float*)d_in[7];
    const float* dec_Whh = (const float*)d_in[8];
    const float* dec_bih = (const float*)d_in[9];
    const float* dec_bhh = (const float*)d_in[10];
    const float* W_av    = (const float*)d_in[11];
    const float* b_av    = (const float*)d_in[12];
    const float* W_cls   = (const float*)d_in[13];
    const float* b_cls   = (const float*)d_in[14];

    float* out = (float*)d_out;
    float* logits_base = out;                          // [S, B, V]
    float* attnw_base  = out + (size_t)SS * BB * VV;   // [S, B, T]

    // ---- workspace carve-up (floats), ~10.2 MB total ----
    float* ws       = (float*)d_ws;
    float* emb_x    = ws;                         // [B*T, E]
    float* gi_enc   = emb_x    + BB * TT * EE;    // [B*T, 3H]
    float* enc_outs = gi_enc   + BB * TT * H3;    // [B, T, H]
    float* h        = enc_outs + BB * TT * HH;    // [BP, H]   padded
    float* x_cat    = h        + BP * HH;         // [BP, E+A] padded
    float* gi_dec   = x_cat    + BP * (EE + AA);  // [BP, 3H]
    float* gh_buf   = gi_dec   + BP * H3;         // [BP, 3H]
    float* merged   = gh_buf   + BP * H3;         // [BP, 2H]  padded
    float* attn_vec = merged   + BP * 2 * HH;     // [BP, A]   padded
    int*   ids      = (int*)(attn_vec + BP * AA); // [B]

    // ---- init padded state ----
    init_state_kernel<<<(BP * 2 * HH + 255) / 256, 256, 0, stream>>>(
        h, x_cat, merged, attn_vec);

    // ---- encoder embedding gather ----
    enc_embed_kernel<<<(BB * TT * EE + 255) / 256, 256, 0, stream>>>(
        inputs, enc_emb, emb_x);

    // ---- hoisted encoder input-gate GEMM: [512,512] x [3072,512]^T ----
    {
        int waves = (BB * TT / 16) * (H3 / 16);   // 6144
        wmma_gemm_xwt<<<(waves + 7) / 8, 256, 0, stream>>>(
            emb_x, enc_Wih, enc_bih, gi_enc, BB * TT, H3, EE, 0);
    }

    // ---- sequential encoder scan: gh matvec + gate per step ----
    for (int t = 0; t < TT; ++t) {
        int waves = H3 / 16;   // 192
        wmma_gemm_xwt<<<(waves + 7) / 8, 256, 0, stream>>>(
            h, enc_Whh, enc_bhh, gh_buf, BB, H3, HH, 0);
        gru_gate_kernel<<<(BB * HH + 255) / 256, 256, 0, stream>>>(
            gi_enc + (size_t)t * H3, TT * H3,
            gh_buf, H3,
            h,
            enc_outs + (size_t)t * HH, TT * HH);
    }
    // h now holds enc_ht; x_cat/attn_vec start at zero (dec_emb[0] == 0).

    // ---- sequential decoder: 512 greedy steps, 5 kernels each ----
    for (int s = 0; s < SS; ++s) {
        // gi = x_cat @ dec_Wih^T + b ; gh = h @ dec_Whh^T + b  (one launch)
        {
            int waves = 2 * (H3 / 16);   // 384
            wmma_gemm_dual<<<(waves + 7) / 8, 256, 0, stream>>>(
                x_cat, dec_Wih, dec_bih, gi_dec,
                h,     dec_Whh, dec_bhh, gh_buf,
                BB, H3, EE + AA);
        }
        // gate -> h & merged[:,H:], then attention -> weights + merged[:,0:H]
        gate_attention_kernel<<<BB, 1024, 0, stream>>>(
            gi_dec, gh_buf, h, inputs, enc_outs, merged,
            attnw_base + (size_t)s * BB * TT);
        // attn_vec = tanh(merged @ W_av^T + b_av)
        {
            int waves = AA / 16;   // 32
            wmma_gemm_xwt<<<(waves + 7) / 8, 256, 0, stream>>>(
                merged, W_av, b_av, attn_vec, BB, AA, 2 * HH, 1);
        }
        // logits = attn_vec @ W_cls^T + b_cls  -> straight into d_out
        {
            int waves = VV / 16;   // 1000
            wmma_gemm_xwt<<<(waves + 7) / 8, 256, 0, stream>>>(
                attn_vec, W_cls, b_cls,
                logits_base + (size_t)s * BB * VV, BB, VV, EE, 0);
        }
        // argmax + build next x_cat
        argmax_xcat_kernel<<<BB, 256, 0, stream>>>(
            logits_base + (size_t)s * BB * VV, dec_emb, attn_vec, x_cat, ids);
    }
}